// Quantize_90787018703333
// MI455X (gfx1250) — compile-verified
//
#include <hip/hip_runtime.h>
#include <hip/hip_bf16.h>

// VQ-VAE quantize for MI455X (gfx1250, wave32).
// N = 65536 rows, D = 64, K = 1024 codes.
//
// d_out layout (concatenated outputs, float32):
//   [0 .. 4194303]  quantized  (65536 x 64)
//   [4194304]       loss (scalar)
//   [4194305 .. ]   encoding indices (65536, stored as float values)

typedef __attribute__((ext_vector_type(2))) float v2f;
typedef __attribute__((ext_vector_type(8))) float v8f;

#define VQ_N 65536
#define VQ_D 64
#define VQ_K 1024

// ---------------------------------------------------------------------------
// Kernel 1: per-code squared norms ||e_k||^2 -> workspace; zero the loss slot.
// ---------------------------------------------------------------------------
__global__ __launch_bounds__(256) void vq_norms_kernel(
    const float* __restrict__ E, float* __restrict__ norms2,
    float* __restrict__ loss) {
  int k = blockIdx.x * 256 + threadIdx.x;  // 4 blocks x 256 = 1024
  float s = 0.0f;
#pragma unroll 8
  for (int d = 0; d < VQ_D; ++d) {
    float e = E[d * VQ_K + k];
    s += e * e;
  }
  norms2[k] = s;
  if (k == 0) *loss = 0.0f;
}

// ---------------------------------------------------------------------------
// Kernel 2: distances via V_WMMA_F32_16X16X4_F32 + per-row argmin.
// Workgroup = 8 waves x 16 rows = 128 rows. 512 workgroups total.
// Codebook column-tiles (64 x 16 fp32 = 4 KB) double-buffered in LDS and
// shared by all 8 waves; global load for tile c+1 overlaps WMMA on tile c.
// B fragments are software-pipelined at depth 2 so each WMMA waits on
// dscnt<=2 rather than a just-issued DS load.
// ---------------------------------------------------------------------------
__global__ __launch_bounds__(256) void vq_argmin_kernel(
    const float* __restrict__ x, const float* __restrict__ E,
    const float* __restrict__ norms2, float* __restrict__ idx_out) {
  __shared__ float btile[2][VQ_D * 16];  // 2 x 4 KB ping-pong, [d][n]

  const int lane = threadIdx.x & 31;
  const int wave = threadIdx.x >> 5;
  const int half = lane >> 4;   // 0: lanes 0-15, 1: lanes 16-31
  const int l16  = lane & 15;
  const int rowbase = blockIdx.x * 128 + wave * 16;
  const int myrow   = rowbase + l16;

  // Cooperative-load geometry: thread t owns (d = t>>2, nq = (t&3)*4).
  const int cld = threadIdx.x >> 2;
  const int clq = (threadIdx.x & 3) * 4;
  const float* esrc = E + cld * VQ_K + clq;  // advance by 16 per column tile
  const int ldst = cld * 16 + clq;

  // A fragments (16x4 fp32 per kstep): lane holds M = l16,
  // VGPR0 = K = 4s + 2*half, VGPR1 = K = 4s + 2*half + 1.
  // 16 ksteps cover D = 64. Kept resident across the whole c-loop (32 VGPRs).
  v2f afrag[16];
  const float* xrow = x + myrow * VQ_D;
#pragma unroll
  for (int s = 0; s < 16; ++s) {
    afrag[s] = *(const v2f*)(xrow + 4 * s + 2 * half);
  }

  // Per-lane base for B-fragment LDS reads: element (d=2*half, n=l16).
  const int bbase = 2 * half * 16 + l16;

  // C/D layout: VGPR v, lane L -> M = v + 8*half, N = l16.
  float best[8];
  int bestIdx[8];
#pragma unroll
  for (int v = 0; v < 8; ++v) {
    best[v] = 3.0e38f;
    bestIdx[v] = 0;
  }

  // Prologue: stage tile 0 into buffer 0.
  {
    float4 t0 = *(const float4*)(esrc);
    *(float4*)(&btile[0][ldst]) = t0;
  }
  __syncthreads();

  for (int c = 0; c < VQ_K / 16; ++c) {
    const int p = c & 1;
    const float* bt = &btile[p][bbase];

    // Issue next tile's global load early; latency hides under the WMMAs.
    float4 nxt;
    if (c < VQ_K / 16 - 1) {
      nxt = *(const float4*)(esrc + (c + 1) * 16);
    }

    float nrm = norms2[c * 16 + l16];

    // B fragment for kstep s: VGPR0 = K-row (4s + 2*half), VGPR1 = next row.
    // Depth-2 pipeline: ds_load for stage s+2 issues before wmma consumes
    // stage s, keeping only 3 fragment registers live.
    v2f b0, b1;
    b0.x = bt[0 * 64];            // d = 4*0 + 2*half
    b0.y = bt[0 * 64 + 16];
    b1.x = bt[1 * 64];            // d = 4*1 + 2*half
    b1.y = bt[1 * 64 + 16];

    v8f acc = {};
#pragma unroll
    for (int s = 0; s < 16; ++s) {
      v2f bn;
      if (s < 14) {
        bn.x = bt[(s + 2) * 64];
        bn.y = bt[(s + 2) * 64 + 16];
      }
      acc = __builtin_amdgcn_wmma_f32_16x16x4_f32(
          /*neg_a=*/false, afrag[s], /*neg_b=*/false, b0,
          /*c_mod=*/(short)0, acc, /*reuse_a=*/false, /*reuse_b=*/false);
      b0 = b1;
      b1 = bn;
    }

    // dist = ||e||^2 - 2 x.e  (||x||^2 dropped: constant per row)
    int col = c * 16 + l16;
#pragma unroll
    for (int v = 0; v < 8; ++v) {
      float dist = nrm - 2.0f * acc[v];
      if (dist < best[v]) {  // c ascending => index ties auto-resolve low
        best[v] = dist;
        bestIdx[v] = col;
      }
    }

    // Stage next tile into the other buffer; it was last read at c-1, which
    // the barrier below (executed at end of c-1) already fenced.
    if (c < VQ_K / 16 - 1) {
      *(float4*)(&btile[1 - p][ldst]) = nxt;
    }
    __syncthreads();
  }

  // Min-reduce across the 16 lanes holding the same row (N dimension).
  // XOR masks 1,2,4,8 never cross the half boundary (bit 4).
#pragma unroll
  for (int off = 1; off < 16; off <<= 1) {
#pragma unroll
    for (int v = 0; v < 8; ++v) {
      float ob = __shfl_xor(best[v], off, 32);
      int oi = __shfl_xor(bestIdx[v], off, 32);
      if (ob < best[v] || (ob == best[v] && oi < bestIdx[v])) {
        best[v] = ob;
        bestIdx[v] = oi;
      }
    }
  }

  if (l16 == 0) {
    // lane 0 holds rows rowbase+0..7 (v=0..7), lane 16 rows rowbase+8..15.
#pragma unroll
    for (int v = 0; v < 8; ++v) {
      idx_out[rowbase + v + 8 * half] = (float)bestIdx[v];
    }
  }
}

// ---------------------------------------------------------------------------
// Kernel 3: gather codewords -> quantized, accumulate 1.25 * MSE into loss.
// One thread per (row, 4 dims): 65536*16 threads = 4096 blocks of 256.
// ---------------------------------------------------------------------------
__global__ __launch_bounds__(256) void vq_gather_kernel(
    const float* __restrict__ x, const float* __restrict__ E,
    const float* __restrict__ idx_f, float* __restrict__ quant,
    float* __restrict__ loss) {
  int gid = blockIdx.x * 256 + threadIdx.x;
  int row = gid >> 4;
  int dq = (gid & 15) * 4;
  int k = (int)idx_f[row];

  float4 xi = *(const float4*)(x + row * VQ_D + dq);
  float q0 = E[(dq + 0) * VQ_K + k];
  float q1 = E[(dq + 1) * VQ_K + k];
  float q2 = E[(dq + 2) * VQ_K + k];
  float q3 = E[(dq + 3) * VQ_K + k];
  float4 q = {q0, q1, q2, q3};
  *(float4*)(quant + row * VQ_D + dq) = q;

  float e0 = q0 - xi.x, e1 = q1 - xi.y, e2 = q2 - xi.z, e3 = q3 - xi.w;
  float ss = e0 * e0 + e1 * e1 + e2 * e2 + e3 * e3;

  __shared__ float red[256];
  red[threadIdx.x] = ss;
  __syncthreads();
#pragma unroll
  for (int st = 128; st > 0; st >>= 1) {
    if (threadIdx.x < st) red[threadIdx.x] += red[threadIdx.x + st];
    __syncthreads();
  }
  if (threadIdx.x == 0) {
    // loss = (e_latent + BETA*q_latent) = 1.25 * mean((q-x)^2)
    atomicAdd(loss, red[0] * (1.25f / (float)(VQ_N * VQ_D)));
  }
}

// ---------------------------------------------------------------------------
extern "C" void kernel_launch(void* const* d_in, const int* in_sizes, int n_in,
                              void* d_out, int out_size, void* d_ws,
                              size_t ws_size, hipStream_t stream) {
  const float* x = (const float*)d_in[0];  // [64,32,32,64]
  const float* E = (const float*)d_in[1];  // [64,1024]

  float* quant = (float*)d_out;            // 4194304 floats
  float* loss = quant + (VQ_N * VQ_D);     // 1 float
  float* idx_out = loss + 1;               // 65536 floats

  float* norms2 = (float*)d_ws;            // 1024 floats scratch

  vq_norms_kernel<<<VQ_K / 256, 256, 0, stream>>>(E, norms2, loss);
  vq_argmin_kernel<<<VQ_N / 128, 256, 0, stream>>>(x, E, norms2, idx_out);
  vq_gather_kernel<<<(VQ_N * 16) / 256, 256, 0, stream>>>(x, E, idx_out, quant,
                                                          loss);
}